// GaussianPooling_68822555951688
// MI455X (gfx1250) — compile-verified
//
#include <hip/hip_runtime.h>

typedef __attribute__((ext_vector_type(2))) float v2f;
typedef __attribute__((ext_vector_type(8))) float v8f;

#define C_DIM 256
#define H_DIM 256
#define W_DIM 256
#define HALF  2           // KERNEL_SIZE 5
#define INV2SIG2 (1.0f / 8.0f)   // 1/(2*sigma^2), sigma=2

// ---------------------------------------------------------------------------
// Phase 1: transpose (C, H*W) -> (H*W, C) through LDS (bank-conflict-free).
// After this, all 256 channels of one pixel are contiguous (1 KB).
// ---------------------------------------------------------------------------
__global__ __launch_bounds__(256) void gp_transpose_chw_to_hwc(
    const float* __restrict__ fm, float* __restrict__ fmT) {
  __shared__ float tile[32][33];
  const int p0 = blockIdx.x * 32;   // pixel base (h*W + w); 32 | W so rows never split
  const int c0 = blockIdx.y * 32;   // channel base
  const int tx = threadIdx.x;       // 0..31
  const int ty = threadIdx.y;       // 0..7
#pragma unroll
  for (int i = 0; i < 4; ++i) {
    const int c = c0 + ty + i * 8;
    tile[ty + i * 8][tx] = fm[(size_t)c * (H_DIM * W_DIM) + p0 + tx];
  }
  __syncthreads();
#pragma unroll
  for (int i = 0; i < 4; ++i) {
    const int p = p0 + ty + i * 8;
    fmT[(size_t)p * C_DIM + c0 + tx] = tile[tx][ty + i * 8];
  }
}

// ---------------------------------------------------------------------------
// Phase 2: one wave = (one keypoint, 16 channels).
// out[c] = sum_{t<25} w[t] * fmT[(y+dy_t)*W + (x+dx_t)][c]
// as D(16x16) = A(16x4) x B(4x16) + C, looped over 7 K-chunks (25 taps -> 28).
// A layout (f32 16x4): lanes 0-15 -> M=lane, VGPR0:K=0 VGPR1:K=1;
//                      lanes 16-31 -> M=lane-16, VGPR0:K=2 VGPR1:K=3.
// B layout (f32 4x16): VGPR0: lanes0-15 row K=0, lanes16-31 row K=2;
//                      VGPR1: lanes0-15 row K=1, lanes16-31 row K=3; N = lane&15.
// D row M=0 lives in acc[0] of lanes 0-15 (N = lane).
// ---------------------------------------------------------------------------
__global__ __launch_bounds__(256) void gp_gather_wmma(
    const float* __restrict__ fmT, const int* __restrict__ kp,
    float* __restrict__ out, int N) {
  const int lane = threadIdx.x & 31;
  const int wid  = blockIdx.x * 8 + (threadIdx.x >> 5);
  const int n    = wid >> 4;
  const int c0   = (wid & 15) << 4;
  if (n >= N) return;                       // uniform per wave -> EXEC all-1 below

  int xi = kp[2 * n + 0];
  int yi = kp[2 * n + 1];
  xi = min(max(xi, HALF), W_DIM - HALF - 1);
  yi = min(max(yi, HALF), H_DIM - HALF - 1);

  // Gaussian weights (separable), normalized by total sum = s*s.
  float g[5]; float s = 0.0f;
#pragma unroll
  for (int i = 0; i < 5; ++i) { float d = (float)(i - 2); g[i] = expf(-d * d * INV2SIG2); s += g[i]; }
  const float inv = 1.0f / (s * s);
  float wv[28];
#pragma unroll
  for (int t = 0; t < 28; ++t) wv[t] = (t < 25) ? g[t / 5] * g[t % 5] * inv : 0.0f;

  const bool lo   = lane < 16;
  const int  col  = c0 + (lane & 15);

  v8f acc = {};
#pragma unroll
  for (int ch = 0; ch < 7; ++ch) {
    const int base = ch * 4;
    // A: broadcast kernel weights per the 16x4 layout
    v2f a;
    a.x = lo ? wv[base + 0] : wv[base + 2];
    a.y = lo ? wv[base + 1] : wv[base + 3];
    // B: patch data, taps t0 = base + (lo?0:2), t1 = t0+1
    const int t0 = base + (lo ? 0 : 2);
    const int t1 = t0 + 1;
    v2f b;
    {
      const int dy = t0 / 5 - 2, dx = t0 % 5 - 2;
      const size_t addr = ((size_t)(yi + dy) * W_DIM + (size_t)(xi + dx)) * C_DIM + col;
      b.x = (t0 < 25) ? fmT[addr] : 0.0f;
    }
    {
      const int dy = t1 / 5 - 2, dx = t1 % 5 - 2;
      const size_t addr = ((size_t)(yi + dy) * W_DIM + (size_t)(xi + dx)) * C_DIM + col;
      b.y = (t1 < 25) ? fmT[addr] : 0.0f;
    }
    // D = A*B + acc  (exact f32 WMMA)
    acc = __builtin_amdgcn_wmma_f32_16x16x4_f32(
        /*neg_a=*/false, a, /*neg_b=*/false, b,
        /*c_mod=*/(short)0, acc, /*reuse_a=*/false, /*reuse_b=*/false);
  }

  if (lane < 16) out[(size_t)n * C_DIM + c0 + lane] = acc[0];  // M=0 row
}

// ---------------------------------------------------------------------------
// Fallback (workspace too small): direct per-(n,c) VALU gather on (C,H,W).
// ---------------------------------------------------------------------------
__global__ __launch_bounds__(256) void gp_gather_direct(
    const float* __restrict__ fm, const int* __restrict__ kp,
    float* __restrict__ out, int N) {
  const int tid = blockIdx.x * 256 + threadIdx.x;
  if (tid >= N * C_DIM) return;
  const int n = tid >> 8;
  const int c = tid & 255;

  int xi = kp[2 * n + 0];
  int yi = kp[2 * n + 1];
  xi = min(max(xi, HALF), W_DIM - HALF - 1);
  yi = min(max(yi, HALF), H_DIM - HALF - 1);

  float g[5]; float s = 0.0f;
#pragma unroll
  for (int i = 0; i < 5; ++i) { float d = (float)(i - 2); g[i] = expf(-d * d * INV2SIG2); s += g[i]; }
  const float inv = 1.0f / (s * s);

  const float* base = fm + (size_t)c * (H_DIM * W_DIM);
  float acc = 0.0f;
#pragma unroll
  for (int i = 0; i < 5; ++i) {
#pragma unroll
    for (int j = 0; j < 5; ++j) {
      acc = fmaf(g[i] * g[j] * inv,
                 base[(size_t)(yi + i - 2) * W_DIM + (xi + j - 2)], acc);
    }
  }
  out[(size_t)n * C_DIM + c] = acc;
}

// ---------------------------------------------------------------------------
extern "C" void kernel_launch(void* const* d_in, const int* in_sizes, int n_in,
                              void* d_out, int out_size, void* d_ws, size_t ws_size,
                              hipStream_t stream) {
  const float* fm  = (const float*)d_in[0];
  const int*   kp  = (const int*)d_in[1];
  float*       out = (float*)d_out;
  const int N = in_sizes[1] / 2;

  const size_t fmT_bytes = (size_t)C_DIM * H_DIM * W_DIM * sizeof(float);
  if (ws_size >= fmT_bytes) {
    float* fmT = (float*)d_ws;
    dim3 tblk(32, 8);
    dim3 tgrd((H_DIM * W_DIM) / 32, C_DIM / 32);
    gp_transpose_chw_to_hwc<<<tgrd, tblk, 0, stream>>>(fm, fmT);

    const int waves  = N * (C_DIM / 16);
    const int blocks = (waves + 7) / 8;
    gp_gather_wmma<<<blocks, 256, 0, stream>>>(fmT, kp, out, N);
  } else {
    const int total  = N * C_DIM;
    const int blocks = (total + 255) / 256;
    gp_gather_direct<<<blocks, 256, 0, stream>>>(fm, kp, out, N);
  }
}